// GPoseT_11304353923105
// MI455X (gfx1250) — compile-verified
//
#include <hip/hip_runtime.h>
#include <hip/hip_bf16.h>
#include <stdint.h>

// Problem constants (fixed by the reference).
#define T_ 256
#define M_ 2
#define K_ 28
#define H_ 64
#define W_ 64
// 3*sigma = 1.8, 2*sigma^2 = 0.72 (exact f32 roundings of the Python doubles)
#define THREE_SIG 1.8f
#define TWO_SIG2  0.72f
#define EPS_ 1e-4f

#define AS1 __attribute__((address_space(1)))
#define AS3 __attribute__((address_space(3)))

typedef int v4i __attribute__((ext_vector_type(4)));

// ---------------------------------------------------------------------------
// Kernel 1: zero-fill 117.4 MB of output at full HBM rate.
// Preferred path: CDNA5 async LDS->global stores (GLOBAL_STORE_ASYNC_FROM_LDS_B128,
// tracked via ASYNCcnt). Each lane streams its own 16B of pre-zeroed LDS; a wave
// issues 512B per instruction, fully coalesced. Fallbacks keep compilation safe.
// ---------------------------------------------------------------------------
__global__ __launch_bounds__(256) void gposet_zero(float4* __restrict__ out, int n4) {
    const int tid    = blockIdx.x * 256 + threadIdx.x;
    const int stride = gridDim.x * 256;

#if defined(__HIP_DEVICE_COMPILE__) && defined(__gfx1250__)
    __shared__ __align__(16) float zbuf[128];   // 32 lanes * 16B of zeros
    if (threadIdx.x < 128) zbuf[threadIdx.x] = 0.0f;
    __syncthreads();                            // DS writes visible before async reads
    float* lsrc_g = &zbuf[(threadIdx.x & 31) * 4];

#if __has_builtin(__builtin_amdgcn_global_store_async_from_lds_b128)
    AS3 v4i* lsrc = (AS3 v4i*)lsrc_g;
    for (int i = tid; i < n4; i += stride) {
        __builtin_amdgcn_global_store_async_from_lds_b128(
            (AS1 v4i*)(out + i), lsrc, /*imm offset*/0, /*cpol*/0);
    }
#else
    // Inline-asm tier: VGLOBAL encoding, VADDR = 64-bit global addr pair,
    // VSRC = LDS byte address, SADDR = off.
    uint32_t lds_addr = (uint32_t)(uintptr_t)lsrc_g;  // LDS segment byte offset
    for (int i = tid; i < n4; i += stride) {
        uint64_t gaddr = (uint64_t)(uintptr_t)(out + i);
        asm volatile("global_store_async_from_lds_b128 %0, %1, off"
                     :: "v"(gaddr), "v"(lds_addr) : "memory");
    }
#endif

#if __has_builtin(__builtin_amdgcn_s_wait_asynccnt)
    __builtin_amdgcn_s_wait_asynccnt(0);
#else
    asm volatile("s_wait_asynccnt 0" ::: "memory");
#endif

#else  // host pass / non-gfx1250: plain 128-bit stores
    const float4 z = make_float4(0.f, 0.f, 0.f, 0.f);
    for (int i = tid; i < n4; i += stride) out[i] = z;
#endif
}

// ---------------------------------------------------------------------------
// Kernel 2: sparse Gaussian splat. One wave (32 lanes) per keypoint (t,m,k);
// lanes 0..24 cover the <=5x5 candidate bbox. Non-negative patch values are
// max-combined via u32 atomic max on the IEEE bit pattern (order-preserving
// for non-negative floats; buffer pre-zeroed to +0.0). Deterministic: max is
// associative + commutative.
// ---------------------------------------------------------------------------
__global__ __launch_bounds__(256) void gposet_splat(const float* __restrict__ kps,
                                                    const float* __restrict__ mv,
                                                    unsigned int* __restrict__ out,
                                                    int nkp) {
    const int g    = blockIdx.x * 256 + threadIdx.x;
    const int kp   = g >> 5;     // keypoint id: uniform across a wave32
    const int lane = g & 31;
    if (kp >= nkp) return;

    const float val = mv[kp];
    if (val < EPS_) return;      // reference drops the whole patch
    if (lane >= 25) return;

    const float mux = kps[kp * 2 + 0];
    const float muy = kps[kp * 2 + 1];

    // bbox exactly as the reference computes it
    const float x0  = floorf(mux - THREE_SIG);
    const float y0  = floorf(muy - THREE_SIG);
    const float edx = fminf(floorf(mux + THREE_SIG) + 1.0f, (float)W_);
    const float edy = fminf(floorf(muy + THREE_SIG) + 1.0f, (float)H_);

    const float xf = x0 + (float)(lane % 5);
    const float yf = y0 + (float)(lane / 5);
    // st = max(x0, 0): since xf >= x0 always, the lower-bound check is xf >= 0.
    if (xf < 0.0f || yf < 0.0f || xf >= edx || yf >= edy) return;

    const float dx = xf - mux;
    const float dy = yf - muy;
    const float d2 = dx * dx + dy * dy;
    const float patch = expf(-d2 / TWO_SIG2) * val;

    const int xi = (int)xf;
    const int yi = (int)yf;
    const int t  = kp / (M_ * K_);
    const int k  = (kp - t * (M_ * K_)) % K_;   // channel; m only selects source

    // out[t][h][w][k], channel-last
    const unsigned int idx =
        (((unsigned)t * H_ + (unsigned)yi) * W_ + (unsigned)xi) * K_ + (unsigned)k;

    atomicMax(out + idx, __float_as_uint(patch));   // global_atomic_max_u32
}

// ---------------------------------------------------------------------------
// kernel_launch
//   d_in[0] = kps        [T,M,K,2] f32
//   d_in[1] = max_values [T,M,K]   f32
//   d_in[2] = img_h (=64), d_in[3] = img_w (=64)  (compile-time constants)
//   d_out   = [T,H,W,K] f32
// ---------------------------------------------------------------------------
extern "C" void kernel_launch(void* const* d_in, const int* in_sizes, int n_in,
                              void* d_out, int out_size, void* d_ws, size_t ws_size,
                              hipStream_t stream) {
    (void)n_in; (void)d_ws; (void)ws_size;

    const float* kps = (const float*)d_in[0];
    const float* mv  = (const float*)d_in[1];

    // Zero fill: out_size = 29,360,128 floats -> 7,340,032 float4s.
    const int n4 = out_size / 4;
    int zero_blocks = (n4 / 4 + 255) / 256;       // ~4 float4 stores per thread
    if (zero_blocks < 1) zero_blocks = 1;
    gposet_zero<<<zero_blocks, 256, 0, stream>>>((float4*)d_out, n4);

    // Scatter: one wave32 per keypoint.
    const int nkp = in_sizes[1];                  // T*M*K = 14336
    const int splat_blocks = (nkp * 32 + 255) / 256;
    gposet_splat<<<splat_blocks, 256, 0, stream>>>(kps, mv, (unsigned int*)d_out, nkp);
}